// SupConLoss_76287209111760
// MI455X (gfx1250) — compile-verified
//
#include <hip/hip_runtime.h>
#include <hip/hip_bf16.h>

typedef __attribute__((ext_vector_type(2))) float v2f;
typedef __attribute__((ext_vector_type(8))) float v8f;
typedef __attribute__((ext_vector_type(4))) unsigned int v4u;
typedef __attribute__((ext_vector_type(4))) int v4i;
typedef __attribute__((ext_vector_type(8))) int v8i;

#define Bc   32
#define Vc   200
#define Dc   256
#define Nc   (Bc * Vc)        // 6400 rows
#define NT   (Nc / 16)        // 400 16x16 tiles per dimension
#define NW   8                // waves per block
#define TPW  (NT / NW)        // 50 column tiles per wave
#define APITCH 260            // padded LDS row pitch (floats) -> conflict-free ds_load_b64
#define TILE_BYTES (16 * Dc * 4)   // 16 KB of tensor data per tile
#define INV_T 14.2857142857142857f
#define POSC  199.0f          // positives per row = V-1

// ---------------- Kernel 1: L2-normalize + permute to contrast order ----------------
__global__ __launch_bounds__(256) void knorm(const float* __restrict__ f,
                                             float* __restrict__ w) {
    int i = blockIdx.x;          // contrast row 0..6399: (view = i/32, batch = i%32)
    int d = threadIdx.x;         // 0..255
    int batch = i & (Bc - 1);
    int view  = i >> 5;
    float x = f[(size_t)batch * (Vc * Dc) + (size_t)view * Dc + d];
    float ss = x * x;
    for (int m = 16; m; m >>= 1) ss += __shfl_xor(ss, m, 32);
    __shared__ float red[8];
    int lane = threadIdx.x & 31, wv = threadIdx.x >> 5;
    if (lane == 0) red[wv] = ss;
    __syncthreads();
    float tot = 0.f;
#pragma unroll
    for (int k = 0; k < 8; k++) tot += red[k];
    float nrm = fmaxf(sqrtf(tot), 1e-12f);
    w[(size_t)i * Dc + d] = x / nrm;
}

// -------- TDM: issue one tensor_load_to_lds for a 16-row x 256-f32 tile -------------
// LDS destination gets 4 DWORDs of padding every 256 DWORDs -> 260-float row pitch.
__device__ __forceinline__ void tdm_load_tile(unsigned lds_addr,
                                              unsigned long long gaddr) {
    unsigned lds  = (unsigned)__builtin_amdgcn_readfirstlane((int)lds_addr);
    unsigned galo = (unsigned)__builtin_amdgcn_readfirstlane((int)(gaddr & 0xFFFFFFFFull));
    unsigned gahi = (unsigned)__builtin_amdgcn_readfirstlane((int)(gaddr >> 32));

    v4u g0;
    g0.x = 1u;                                            // count=1, user D#, no gather
    g0.y = lds;                                           // lds_addr (bytes)
    g0.z = galo;                                          // global_addr[31:0]
    g0.w = (gahi & 0x01FFFFFFu) | (2u << 30);             // global_addr[56:32] | type=2

    v8i g1;
    g1[0] = (2 << 16)                                     // data_size = 4 bytes
          | (1 << 20)                                     // pad_enable
          | (7 << 22)                                     // pad_interval = 256 DWORDs
          | (3 << 25);                                    // pad_amount = 4 DWORDs
    g1[1] = (int)(256u << 16);                            // tensor_dim0 = 256 (lo16)
    g1[2] = (int)((unsigned)Nc << 16);                    // tensor_dim0 hi | tensor_dim1 lo
    g1[3] = (int)(256u << 16);                            // tensor_dim1 hi | tile_dim0 = 256
    g1[4] = 16;                                           // tile_dim1 = 16 | tile_dim2 = 0
    g1[5] = 256;                                          // tensor_dim0_stride = 256 elems
    g1[6] = 0;
    g1[7] = 0;

#if __has_builtin(__builtin_amdgcn_tensor_load_to_lds_d2)
    __builtin_amdgcn_tensor_load_to_lds_d2((v4i)g0, g1, 0);
#elif __clang_major__ >= 23
    v4i z4 = {};
    v8i z8 = {};
    __builtin_amdgcn_tensor_load_to_lds(g0, g1, z4, z4, z8, 0);
#else
    v4i z4 = {};
    __builtin_amdgcn_tensor_load_to_lds(g0, g1, z4, z4, 0);
#endif
}

// ---------------- Kernel 2: fused Gram (fp32 WMMA, TDM-fed) + online LSE ------------
__global__ __launch_bounds__(256) void kgram(const float* __restrict__ w,
                                             float* __restrict__ partial) {
    __shared__ float As[16 * APITCH];             // A tile (manually staged)
    __shared__ float Bs[NW][2][16 * APITCH];      // per-wave double-buffered B tiles (TDM)
    __shared__ float sm[16][NW], sz[16][NW], sp[16][NW];
    __shared__ float lr[16];

    const int tid  = threadIdx.x;
    const int ib   = blockIdx.x * 16;
    const int lane = tid & 31;
    const int wv   = tid >> 5;
    const int half = lane >> 4;
    const int l16  = lane & 15;
    const int koff = half * 2;

    // stage A tile (16 x 256 f32) into padded LDS, coalesced
    for (int idx = tid; idx < 16 * Dc; idx += 256) {
        int r = idx >> 8, c = idx & 255;
        As[r * APITCH + c] = w[(size_t)(ib + r) * Dc + c];
    }
    __syncthreads();

    const float* Ap = &As[l16 * APITCH + koff];

    // LDS byte addresses of this wave's two B buffers (low 32 bits of flat = LDS offset)
    unsigned bbase[2];
    bbase[0] = (unsigned)(unsigned long long)(const void*)&Bs[wv][0][0];
    bbase[1] = (unsigned)(unsigned long long)(const void*)&Bs[wv][1][0];
    const unsigned long long gbase = (unsigned long long)w;

    float m[8], z[8], s[8];
#pragma unroll
    for (int r = 0; r < 8; r++) { m[r] = -1e30f; z[r] = 0.f; s[r] = 0.f; }

    // prime the pipeline: tile jt = wv into buffer 0
    tdm_load_tile(bbase[0], gbase + (unsigned long long)wv * TILE_BYTES);

    for (int idx = 0; idx < TPW; ++idx) {        // wave-uniform trip count
        const int jt  = wv + idx * NW;
        const int buf = idx & 1;
        if (idx + 1 < TPW) {
            tdm_load_tile(bbase[buf ^ 1],
                          gbase + (unsigned long long)(jt + NW) * TILE_BYTES);
            __builtin_amdgcn_s_wait_tensorcnt(1);   // oldest (current buf) complete
        } else {
            __builtin_amdgcn_s_wait_tensorcnt(0);
        }
        asm volatile("" ::: "memory");              // TDM wrote LDS: force re-loads

        const float* Bp = &Bs[wv][buf][l16 * APITCH + koff];
        v8f acc = {};
#pragma unroll 16
        for (int k = 0; k < Dc; k += 4) {
            v2f a = *(const v2f*)(Ap + k);          // ds_load_b64 (conflict-free)
            v2f b = *(const v2f*)(Bp + k);          // ds_load_b64 (conflict-free)
            acc = __builtin_amdgcn_wmma_f32_16x16x4_f32(
                false, a, false, b, (short)0, acc, false, false);
        }

        int jj = jt * 16 + l16;                     // global column index
#pragma unroll
        for (int r = 0; r < 8; r++) {
            int   gi   = ib + r + 8 * half;         // global row for this C element
            float raw  = acc[r] * INV_T;
            bool  diag = (jj == gi);
            bool  same = (((jj ^ gi) & (Bc - 1)) == 0);
            float mn = fmaxf(m[r], raw);            // max includes diagonal (as reference)
            float ze = diag ? 0.f : __expf(raw - mn);
            z[r] = z[r] * __expf(m[r] - mn) + ze;
            m[r] = mn;
            s[r] += (same && !diag) ? raw : 0.f;
        }
    }

    // reduce across the 16 lanes holding the same row (N dimension)
#pragma unroll
    for (int r = 0; r < 8; r++) {
        for (int msk = 1; msk < 16; msk <<= 1) {
            float mo = __shfl_xor(m[r], msk, 32);
            float zo = __shfl_xor(z[r], msk, 32);
            float so = __shfl_xor(s[r], msk, 32);
            float mn = fmaxf(m[r], mo);
            z[r] = z[r] * __expf(m[r] - mn) + zo * __expf(mo - mn);
            m[r] = mn;
            s[r] += so;
        }
    }
    if (l16 == 0) {                  // lanes 0 / 16 hold reduced rows 0-7 / 8-15
        int rb = 8 * half;
#pragma unroll
        for (int r = 0; r < 8; r++) {
            sm[rb + r][wv] = m[r];
            sz[rb + r][wv] = z[r];
            sp[rb + r][wv] = s[r];
        }
    }
    __syncthreads();

    if (tid < 16) {                  // merge the 8 wave-states per row
        float M = -1e30f, Z = 0.f, S = 0.f;
#pragma unroll
        for (int q = 0; q < NW; q++) {
            float mo = sm[tid][q], zo = sz[tid][q], so = sp[tid][q];
            float mn = fmaxf(M, mo);
            Z = Z * __expf(M - mn) + zo * __expf(mo - mn);
            M = mn;
            S += so;
        }
        lr[tid] = S / POSC - M - __logf(Z + 1e-8f);
    }
    __syncthreads();
    if (tid == 0) {
        float t = 0.f;
#pragma unroll
        for (int r = 0; r < 16; r++) t += lr[r];
        partial[blockIdx.x] = t;
    }
}

// ---------------- Kernel 3: deterministic final reduction --------------------------
__global__ __launch_bounds__(256) void kfinal(const float* __restrict__ p,
                                              float* __restrict__ out) {
    int tid = threadIdx.x;
    float t = 0.f;
    for (int i = tid; i < NT; i += 256) t += p[i];
    for (int msk = 16; msk; msk >>= 1) t += __shfl_xor(t, msk, 32);
    __shared__ float red[8];
    int lane = tid & 31, wv = tid >> 5;
    if (lane == 0) red[wv] = t;
    __syncthreads();
    if (tid == 0) {
        float tot = 0.f;
#pragma unroll
        for (int i = 0; i < 8; i++) tot += red[i];
        out[0] = -tot / (float)Nc;   // loss = -(T/baseT) * mean, T/baseT == 1
    }
}

extern "C" void kernel_launch(void* const* d_in, const int* in_sizes, int n_in,
                              void* d_out, int out_size, void* d_ws, size_t ws_size,
                              hipStream_t stream) {
    const float* feats = (const float*)d_in[0];
    float* out   = (float*)d_out;
    float* wnorm = (float*)d_ws;                       // 6400*256 f32 = 6.55 MB
    float* parts = wnorm + (size_t)Nc * Dc;            // + 400 f32 partials

    knorm <<<Nc, 256, 0, stream>>>(feats, wnorm);
    kgram <<<NT, 256, 0, stream>>>(wnorm, parts);
    kfinal<<<1,  256, 0, stream>>>(parts, out);
}